// GraphSAGE_31044023615683
// MI455X (gfx1250) — compile-verified
//
#include <hip/hip_runtime.h>
#include <math.h>

#define N_NODES 20000
#define DEGREE  32
#define IN_DIM  128
#define OUT_DIM 128
#define LDS_STRIDE (IN_DIM + 4)   // pad: A-fragment column reads hit distinct banks

#define TILE_M 16
#define WAVES_PER_BLOCK 2

typedef __attribute__((ext_vector_type(2))) float v2f;
typedef __attribute__((ext_vector_type(8))) float v8f;

__global__ __launch_bounds__(WAVES_PER_BLOCK * 32)
void sage_wmma_kernel(const int* __restrict__ nidx,
                      const float* __restrict__ emb,
                      const float* __restrict__ W,
                      float* __restrict__ out)
{
    // Per-wave 16x128 tile: holds neighbor means, then reused for h = relu(means @ W)
    __shared__ float tile[WAVES_PER_BLOCK][TILE_M][LDS_STRIDE];
    __shared__ float inv_norm[WAVES_PER_BLOCK][TILE_M];

    const int lane  = threadIdx.x & 31;
    const int wave  = threadIdx.x >> 5;
    const int node0 = (blockIdx.x * WAVES_PER_BLOCK + wave) * TILE_M;
    if (node0 >= N_NODES) return;

    // ---------------- Phase A: neighbor mean (coalesced row gather) ------------
    // Lane owns float4 dim-chunk [4*lane, 4*lane+4): a wave reads one full 512B
    // embedding row per neighbor, perfectly coalesced, L2-resident.
    const float inv_deg = 1.0f / (float)DEGREE;
    for (int m = 0; m < TILE_M; ++m) {
        const int* __restrict__ nb = nidx + (size_t)(node0 + m) * DEGREE;
        float ax = 0.f, ay = 0.f, az = 0.f, aw = 0.f;
        #pragma unroll 4
        for (int e = 0; e < DEGREE; ++e) {
            const int idx = nb[e];
            const float4 v =
                *(const float4*)(emb + (size_t)idx * IN_DIM + 4 * lane);
            ax += v.x; ay += v.y; az += v.z; aw += v.w;
        }
        float4 r;
        r.x = ax * inv_deg; r.y = ay * inv_deg;
        r.z = az * inv_deg; r.w = aw * inv_deg;
        *(float4*)&tile[wave][m][4 * lane] = r;
    }
    __syncthreads();

    // ---------------- Phase B: h-tile = means @ W via V_WMMA_F32_16X16X4_F32 ---
    // A (16x4 f32) layout: lanes 0-15 hold K=k0,k0+1; lanes 16-31 hold K=k0+2,k0+3.
    // B (4x16 f32) mirrors A with N across lanes. 32 K-steps x 8 N-tiles.
    const int row  = lane & 15;
    const int ksel = (lane < 16) ? 0 : 2;

    v8f c[8] = {};   // 8 N-tiles of 16 cols -> full 128 output dims
    #pragma unroll 4
    for (int kk = 0; kk < IN_DIM / 4; ++kk) {
        const int k0 = 4 * kk + ksel;
        v2f a;
        a.x = tile[wave][row][k0];
        a.y = tile[wave][row][k0 + 1];
        #pragma unroll
        for (int n = 0; n < 8; ++n) {
            v2f b;
            b.x = W[(size_t)k0 * OUT_DIM + n * 16 + row];
            b.y = W[(size_t)(k0 + 1) * OUT_DIM + n * 16 + row];
            c[n] = __builtin_amdgcn_wmma_f32_16x16x4_f32(
                /*neg_a=*/false, a, /*neg_b=*/false, b,
                /*c_mod=*/(short)0, c[n],
                /*reuse_a=*/false, /*reuse_b=*/false);
        }
    }

    // ---------------- Phase C: ReLU, spill D tiles to LDS (reuse means buf) ----
    // D layout: VGPR r, lanes 0-15 -> (M=r, N=col), lanes 16-31 -> (M=r+8, N=col)
    __syncthreads();   // all A-fragment reads done before overwrite
    const int mbase = (lane < 16) ? 0 : 8;
    #pragma unroll
    for (int n = 0; n < 8; ++n) {
        #pragma unroll
        for (int r = 0; r < 8; ++r) {
            float v = c[n][r];
            v = fmaxf(v, 0.0f);                     // ReLU
            tile[wave][mbase + r][n * 16 + row] = v;
        }
    }
    __syncthreads();

    // ---------------- Phase D: per-row L2 norm (eps = 1e-12) -------------------
    if (lane < TILE_M) {
        float ss = 0.0f;
        #pragma unroll
        for (int d = 0; d < OUT_DIM; d += 4) {
            const float4 v = *(const float4*)&tile[wave][lane][d];
            ss += v.x * v.x + v.y * v.y + v.z * v.z + v.w * v.w;
        }
        const float nrm = fmaxf(sqrtf(ss), 1e-12f);
        inv_norm[wave][lane] = 1.0f / nrm;
    }
    __syncthreads();

    // ---------------- Phase E: scaled coalesced write-out ----------------------
    for (int m = 0; m < TILE_M; ++m) {
        const float s = inv_norm[wave][m];
        const float4 v = *(const float4*)&tile[wave][m][4 * lane];
        float4 o;
        o.x = v.x * s; o.y = v.y * s; o.z = v.z * s; o.w = v.w * s;
        *(float4*)(out + (size_t)(node0 + m) * OUT_DIM + 4 * lane) = o;
    }
}

extern "C" void kernel_launch(void* const* d_in, const int* in_sizes, int n_in,
                              void* d_out, int out_size, void* d_ws, size_t ws_size,
                              hipStream_t stream) {
    const int*   nidx = (const int*)d_in[0];
    // d_in[1] = segment_ids: unused (fixed degree = 32, ids are arange/32)
    const float* emb  = (const float*)d_in[2];
    const float* W    = (const float*)d_in[3];
    float*       out  = (float*)d_out;

    const int nodes_per_block = WAVES_PER_BLOCK * TILE_M;           // 32
    const int grid = (N_NODES + nodes_per_block - 1) / nodes_per_block;  // 625
    sage_wmma_kernel<<<grid, WAVES_PER_BLOCK * 32, 0, stream>>>(nidx, emb, W, out);
}